// Voroloss_81286551044463
// MI455X (gfx1250) — compile-verified
//
#include <hip/hip_runtime.h>
#include <float.h>
#include <math.h>

typedef __attribute__((ext_vector_type(2))) float v2f;
typedef __attribute__((ext_vector_type(8))) float v8f;

#define BATCH   4
#define NPTS    16384
#define MS      4096
#define KNN     11
#define PPB     256     // points per block (one per thread)
#define CHUNK   64      // spoints staged per iteration
#define NCHUNK  (MS / CHUNK)
#define DSTRIDE 66      // padded LDS stride for the dot tile (bank-conflict free)

__global__ __launch_bounds__(PPB)
void voroloss_kernel(const float* __restrict__ points,
                     const float* __restrict__ spoints,
                     float* __restrict__ out)
{
    __shared__ float ppx[PPB], ppy[PPB], ppz[PPB];
    __shared__ float csx[CHUNK], csy[CHUNK], csz[CHUNK], cssq[CHUNK];
    __shared__ float dotT[PPB * DSTRIDE];

    const int b  = blockIdx.x / (NPTS / PPB);
    const int pb = blockIdx.x % (NPTS / PPB);
    const int t  = threadIdx.x;
    const int m0 = pb * PPB;

    const float* __restrict__ P  = points  + (size_t)(b * NPTS + m0) * 3;
    const float* __restrict__ SP = spoints + (size_t)b * MS * 3;

    // Stage this block's 256 points into LDS (and keep own point in regs).
    const float px = P[t * 3 + 0];
    const float py = P[t * 3 + 1];
    const float pz = P[t * 3 + 2];
    ppx[t] = px; ppy[t] = py; ppz[t] = pz;
    const float p2 = px * px + py * py + pz * pz;

    float dk[KNN];
    int   ik[KNN];
#pragma unroll
    for (int k = 0; k < KNN; ++k) { dk[k] = FLT_MAX; ik[k] = 0; }

    const int  w    = t >> 5;        // wave id: 0..7
    const int  lane = t & 31;
    const int  lr   = lane & 15;
    const bool hi   = lane >= 16;

    __syncthreads();

    // A operands are chunk-invariant: build once per row-tile.
    // 32-bit A 16x4 layout: lanes 0-15 hold K=0 (VGPR0) and K=1 (VGPR1),
    // lanes 16-31 hold K=2 (VGPR0) and K=3 (VGPR1). Row = (x,y,z,0).
    // Read all coords unconditionally, then cndmask-select (keeps EXEC full,
    // avoids saveexec-predicated LDS loads).
    v2f aA[2];
#pragma unroll
    for (int rt = 0; rt < 2; ++rt) {
        const int   row0 = w * 32 + rt * 16;
        const float ax = ppx[row0 + lr];
        const float ay = ppy[row0 + lr];
        const float az = ppz[row0 + lr];
        aA[rt].x = hi ? az   : ax;
        aA[rt].y = hi ? 0.0f : ay;
    }

    for (int c = 0; c < NCHUNK; ++c) {
        // Stage chunk of spoints (waves 0-1; wave-uniform -> no intra-wave div).
        if (t < CHUNK) {
            const int j = c * CHUNK + t;
            const float x = SP[j * 3 + 0];
            const float y = SP[j * 3 + 1];
            const float z = SP[j * 3 + 2];
            csx[t] = x; csy[t] = y; csz[t] = z;
            cssq[t] = x * x + y * y + z * z;
            // Warm L0/L2 for the next chunk (global_prefetch_b8).
            if (c + 1 < NCHUNK)
                __builtin_prefetch(SP + (size_t)(c * CHUNK + CHUNK + t) * 3, 0, 0);
        }
        __syncthreads();

        // 256x64 dot tile: 16 row-tiles x 4 col-tiles of 16x16 WMMA;
        // each of the 8 waves does its 2 row-tiles x 4 col-tiles.
#pragma unroll
        for (int rt = 0; rt < 2; ++rt) {
            const int row0 = w * 32 + rt * 16;
#pragma unroll
            for (int ct = 0; ct < CHUNK / 16; ++ct) {
                const int c0 = ct * 16;
                // B 4x16 layout: VGPR0 = rows K=0 (lanes 0-15) / K=2 (lanes 16-31),
                //                VGPR1 = rows K=1 / K=3. Column = (x,y,z,0).
                const float bx = csx[c0 + lr];
                const float by = csy[c0 + lr];
                const float bz = csz[c0 + lr];
                v2f bb;
                bb.x = hi ? bz   : bx;
                bb.y = hi ? 0.0f : by;

                v8f acc = {0.f, 0.f, 0.f, 0.f, 0.f, 0.f, 0.f, 0.f};
                acc = __builtin_amdgcn_wmma_f32_16x16x4_f32(
                    /*neg_a=*/false, aA[rt], /*neg_b=*/false, bb,
                    /*c_mod=*/(short)0, acc, /*reuse_a=*/false, /*reuse_b=*/false);

                // D layout: VGPR r -> row (r | r+8), col = lane%16.
                const int col   = c0 + lr;
                const int rbase = row0 + (hi ? 8 : 0);
#pragma unroll
                for (int r = 0; r < 8; ++r)
                    dotT[(rbase + r) * DSTRIDE + col] = acc[r];
            }
        }
        __syncthreads();

        // Each thread scans its own point's row; maintain sorted top-11.
        const int cb = c * CHUNK;
#pragma unroll 4
        for (int j = 0; j < CHUNK; ++j) {
            const float d  = p2 + cssq[j] - 2.0f * dotT[t * DSTRIDE + j];
            const int   id = cb + j;
            if (d < dk[KNN - 1]) {
                float cd = d; int ci = id;
#pragma unroll
                for (int k = 0; k < KNN; ++k) {
                    const bool  lt = cd < dk[k];
                    const float td = dk[k];
                    const int   ti = ik[k];
                    dk[k] = lt ? cd : td;  ik[k] = lt ? ci : ti;
                    cd    = lt ? td : cd;  ci    = lt ? ti : ci;
                }
            }
        }
        __syncthreads();
    }

    // Voronoi-edge loss from the sorted top-11 (dk[0] = nearest center).
    const int   i0  = ik[0];
    const float s0x = SP[i0 * 3 + 0];
    const float s0y = SP[i0 * 3 + 1];
    const float s0z = SP[i0 * 3 + 2];
    const float cx = px - s0x, cy = py - s0y, cz = pz - s0z;

    float best = FLT_MAX;
#pragma unroll
    for (int k = 1; k < KNN; ++k) {
        const int ii = ik[k];
        const float ex = SP[ii * 3 + 0] - s0x;
        const float ey = SP[ii * 3 + 1] - s0y;
        const float ez = SP[ii * 3 + 2] - s0z;
        const float el = sqrtf(ex * ex + ey * ey + ez * ez);
        const float vl = (cx * ex + cy * ey + cz * ez) / el;
        const float q  = vl - 0.5f * el;
        best = fminf(best, q * q);
    }
    out[(size_t)b * NPTS + m0 + t] = best;
}

extern "C" void kernel_launch(void* const* d_in, const int* in_sizes, int n_in,
                              void* d_out, int out_size, void* d_ws, size_t ws_size,
                              hipStream_t stream) {
    const float* points  = (const float*)d_in[0];   // (4, 16384, 3) f32
    const float* spoints = (const float*)d_in[1];   // (4, 4096, 3)  f32
    float* out = (float*)d_out;                     // (4, 16384)    f32

    dim3 grid(BATCH * (NPTS / PPB));
    dim3 block(PPB);
    voroloss_kernel<<<grid, block, 0, stream>>>(points, spoints, out);
}